// MultiHeadAttention_22677427323377
// MI455X (gfx1250) — compile-verified
//
#include <hip/hip_runtime.h>
#include <hip/hip_bf16.h>
#include <stdint.h>

// MI455X / gfx1250, wave32. All matrix math via V_WMMA_F32_16X16X4_F32 (exact fp32).

typedef float v8f __attribute__((ext_vector_type(8)));
typedef float v2f __attribute__((ext_vector_type(2)));

constexpr int kH  = 8;      // heads
constexpr int kB  = 16;     // batch
constexpr int kG  = 1024;   // graph size
constexpr int kD  = 512;    // input dim
constexpr int kKD = 64;     // qkv dim (head dim 8)
constexpr int kDH = 8;      // per-head dim
constexpr int kE  = 512;    // output dim
constexpr int kM  = kB * kG;  // 16384 rows

// ---- CDNA5 async global->LDS path (guarded; falls back to float4 staging) ----
#if defined(__has_builtin)
#if __has_builtin(__builtin_amdgcn_global_load_async_to_lds_b128)
#define HAVE_ASYNC_B128 1
#endif
#if __has_builtin(__builtin_amdgcn_s_wait_asynccnt)
#define HAVE_WAIT_ASYNC 1
#endif
#endif

// The builtin's prototype (from the round-2 diagnostic) is
//   (int4 as1*, int4 as3*, int offset, int cpol)
// where as1 prints as __device__ and as3 as __shared__ in HIP mode.
typedef int v4i_ld __attribute__((vector_size(4 * sizeof(int))));
#define AS1_V4I(p) \
  ((__attribute__((address_space(1))) v4i_ld*)(uintptr_t)(p))
#define AS3_V4I(p) \
  ((__attribute__((address_space(3))) v4i_ld*)(uintptr_t)(p))

static __device__ __forceinline__ void wait_asynccnt0() {
#if defined(HAVE_WAIT_ASYNC)
  __builtin_amdgcn_s_wait_asynccnt(0);
#else
  asm volatile("s_wait_asynccnt 0x0" ::: "memory");
#endif
}

static __device__ __forceinline__ v8f wmma4(v2f a, v2f b, v8f c) {
  // D = A(16x4) * B(4x16) + C(16x16), fp32
  return __builtin_amdgcn_wmma_f32_16x16x4_f32(false, a, false, b, (short)0, c,
                                               false, false);
}

// xor-swizzle reductions over the 16 lanes that hold one matrix row
#define SWZF(x, imm) \
  __int_as_float(__builtin_amdgcn_ds_swizzle(__float_as_int(x), (imm)))

static __device__ __forceinline__ float rowmax16(float x) {
  x = fmaxf(x, SWZF(x, 0x041f));  // xor 1
  x = fmaxf(x, SWZF(x, 0x081f));  // xor 2
  x = fmaxf(x, SWZF(x, 0x101f));  // xor 4
  x = fmaxf(x, SWZF(x, 0x201f));  // xor 8
  return x;
}
static __device__ __forceinline__ float rowsum16(float x) {
  x += SWZF(x, 0x041f);
  x += SWZF(x, 0x081f);
  x += SWZF(x, 0x101f);
  x += SWZF(x, 0x201f);
  return x;
}

// ---------------------------------------------------------------------------
// Kernel 1: Qm/Km/Vm (16384x64 each) = q (16384x512) @ W{q,k,v}^T (512x64)
// ---------------------------------------------------------------------------
constexpr int S1 = 68;  // LDS row stride (floats): conflict-free & 16B aligned

__global__ __launch_bounds__(256) void qkv_proj_kernel(
    const float* __restrict__ q, const float* __restrict__ Wq,
    const float* __restrict__ Wk, const float* __restrict__ Wv,
    float* __restrict__ Qm, float* __restrict__ Km, float* __restrict__ Vm) {
  __shared__ __align__(16) float qs[128 * S1];
  __shared__ __align__(16) float wsh[192 * S1];
  const int t = threadIdx.x;
  const int lane = t & 31;
  const int w = t >> 5;        // wave 0..7 -> rows w*16..
  const int m = lane & 15;     // A/B lane index
  const int hi = lane >> 4;    // k-half selector
  const int row0 = blockIdx.x * 128;

  v8f acc[12];
#pragma unroll
  for (int i = 0; i < 12; ++i)
#pragma unroll
    for (int v = 0; v < 8; ++v) acc[i][v] = 0.0f;

  for (int kc = 0; kc < 8; ++kc) {  // K chunks of 64
    __syncthreads();
    // stage q tile: 128 rows x 64 cols
#pragma unroll
    for (int i = 0; i < 8; ++i) {
      int idx = t + i * 256;  // 0..2047 float4s
      int r = idx >> 4, c4 = (idx & 15) << 2;
      float4 vq = *(const float4*)(q + (size_t)(row0 + r) * kD + kc * 64 + c4);
      *(float4*)(&qs[r * S1 + c4]) = vq;
    }
    // stage Wq|Wk|Wv tiles: 3 x 64 rows x 64 cols
#pragma unroll
    for (int i = 0; i < 12; ++i) {
      int idx = t + i * 256;  // 0..3071 float4s
      int mat = idx >> 10;
      int r = (idx >> 4) & 63, c4 = (idx & 15) << 2;
      const float* Wp = (mat == 0) ? Wq : ((mat == 1) ? Wk : Wv);
      float4 vw = *(const float4*)(Wp + (size_t)r * kD + kc * 64 + c4);
      *(float4*)(&wsh[(mat * 64 + r) * S1 + c4]) = vw;
    }
    // prefetch next q chunk: 128 rows x 2 cachelines (128B) each
    if (kc + 1 < 8) {
      int pr = t >> 1, ph = t & 1;
      __builtin_prefetch(q + (size_t)(row0 + pr) * kD + (kc + 1) * 64 + ph * 32,
                         0, 3);
    }
    __syncthreads();

    for (int ks = 0; ks < 16; ++ks) {  // 16 k-steps of 4
      const int kb = ks * 4 + 2 * hi;
      v2f a;
      a[0] = qs[(w * 16 + m) * S1 + kb + 0];
      a[1] = qs[(w * 16 + m) * S1 + kb + 1];
#pragma unroll
      for (int nt = 0; nt < 12; ++nt) {
        v2f bf;
        bf[0] = wsh[(nt * 16 + m) * S1 + kb + 0];
        bf[1] = wsh[(nt * 16 + m) * S1 + kb + 1];
        acc[nt] = wmma4(a, bf, acc[nt]);
      }
    }
  }
  // store: D layout -> row = v + 8*hi, col = m
#pragma unroll
  for (int nt = 0; nt < 12; ++nt) {
    float* dst = (nt < 4) ? Qm : ((nt < 8) ? Km : Vm);
    const int col = ((nt & 3) << 4) + m;
#pragma unroll
    for (int v = 0; v < 8; ++v) {
      int row = row0 + w * 16 + v + 8 * hi;
      dst[(size_t)row * kKD + col] = acc[nt][v];
    }
  }
}

// ---------------------------------------------------------------------------
// Kernel 2: flash attention per (h,b). Head (h,b) of Q/K/V is the contiguous
// 128x64 block starting at row h*2048+b*128, viewed as (1024 x 8).
// K/V staged to LDS unpadded (stride 8) so the whole 32KB block is a flat,
// 16B-granular copy -> eligible for GLOBAL_LOAD_ASYNC_TO_LDS_B128.
// ---------------------------------------------------------------------------
constexpr int SKV = 8;   // K/V LDS row stride (contiguous)
constexpr int SP = 17;   // P staging row stride

__global__ __launch_bounds__(256) void attn_kernel(
    const float* __restrict__ Qm, const float* __restrict__ Km,
    const float* __restrict__ Vm, const unsigned char* __restrict__ mask,
    float* __restrict__ Hm) {
  __shared__ __align__(16) float Klds[kG * SKV];
  __shared__ __align__(16) float Vlds[kG * SKV];
  __shared__ __align__(16) float Pst[8 * 16 * SP];
  const int t = threadIdx.x, lane = t & 31, w = t >> 5;
  const int n = lane & 15, hi = lane >> 4;
  const int hb = blockIdx.x;
  const int h = hb >> 4, b = hb & 15;
  const size_t base = (size_t)(h * 2048 + b * 128) * kKD;
  const float* Qg = Qm + base;

  // stage K and V (1024 x 8 each = 32KB each) into LDS
#if defined(HAVE_ASYNC_B128)
  {
    const float* Kg = Km + base;
    const float* Vg = Vm + base;
#pragma unroll
    for (int i = 0; i < 8; ++i) {
      int v4 = t + i * 256;  // 0..2047 16B chunks
      __builtin_amdgcn_global_load_async_to_lds_b128(
          AS1_V4I(Kg + (size_t)v4 * 4), AS3_V4I(&Klds[v4 * 4]), 0, 0);
      __builtin_amdgcn_global_load_async_to_lds_b128(
          AS1_V4I(Vg + (size_t)v4 * 4), AS3_V4I(&Vlds[v4 * 4]), 0, 0);
    }
    wait_asynccnt0();
  }
#else
  {
    const float4* K4 = (const float4*)(Km + base);
    const float4* V4 = (const float4*)(Vm + base);
#pragma unroll
    for (int i = 0; i < 8; ++i) {
      int v4 = t + i * 256;  // 0..2047
      *(float4*)(&Klds[v4 * 4]) = K4[v4];
      *(float4*)(&Vlds[v4 * 4]) = V4[v4];
    }
  }
#endif
  __syncthreads();

  float* Pw = &Pst[w * 16 * SP];  // per-wave 16x16 staging tile

  for (int qt = 0; qt < 2; ++qt) {
    const int qbase = blockIdx.y * 256 + (w * 2 + qt) * 16;
    // Q A-fragments for k = 0..3 and 4..7  (A lane index == n)
    v2f aQ[2];
#pragma unroll
    for (int s = 0; s < 2; ++s) {
      aQ[s][0] = Qg[(size_t)(qbase + n) * kDH + s * 4 + 2 * hi + 0];
      aQ[s][1] = Qg[(size_t)(qbase + n) * kDH + s * 4 + 2 * hi + 1];
    }

    float mrow[8], lrow[8];
    v8f O;
#pragma unroll
    for (int v = 0; v < 8; ++v) {
      mrow[v] = -1e30f;
      lrow[v] = 0.0f;
      O[v] = 0.0f;
    }

    for (int kt = 0; kt < 64; ++kt) {
      const int key0 = kt * 16;
      const bool km = mask[b * kG + key0 + n] != 0;

      // S = Q (16x8) @ K^T (8x16)
      v8f S;
      {
        v2f bk0, bk1;
        bk0[0] = Klds[(key0 + n) * SKV + 2 * hi + 0];
        bk0[1] = Klds[(key0 + n) * SKV + 2 * hi + 1];
        bk1[0] = Klds[(key0 + n) * SKV + 4 + 2 * hi + 0];
        bk1[1] = Klds[(key0 + n) * SKV + 4 + 2 * hi + 1];
        v8f z;
#pragma unroll
        for (int v = 0; v < 8; ++v) z[v] = 0.0f;
        S = wmma4(aQ[0], bk0, z);
        S = wmma4(aQ[1], bk1, S);
      }
      // scale + mask (masked scores become exactly -30, like the reference)
#pragma unroll
      for (int v = 0; v < 8; ++v) S[v] = km ? -30.0f : S[v] * 0.125f;

      // online softmax; masked exp(-30-m) stays in the denominator,
      // but is zeroed in P before P@V (matches where(mask,0,attn))
#pragma unroll
      for (int v = 0; v < 8; ++v) {
        float tmax = rowmax16(S[v]);
        float nm = fmaxf(mrow[v], tmax);
        float scale = __expf(mrow[v] - nm);
        float p = __expf(S[v] - nm);
        lrow[v] = lrow[v] * scale + rowsum16(p);
        O[v] *= scale;
        mrow[v] = nm;
        Pw[(v + 8 * hi) * SP + n] = km ? 0.0f : p;  // D-layout -> LDS
      }
      // P (16x16, A-layout via LDS bounce) @ V (16x8 padded to 16)
#pragma unroll
      for (int ts = 0; ts < 4; ++ts) {
        v2f ap, bv;
        ap[0] = Pw[n * SP + ts * 4 + 2 * hi + 0];
        ap[1] = Pw[n * SP + ts * 4 + 2 * hi + 1];
        const int kr = key0 + ts * 4 + 2 * hi;
        bv[0] = (n < 8) ? Vlds[(kr + 0) * SKV + n] : 0.0f;
        bv[1] = (n < 8) ? Vlds[(kr + 1) * SKV + n] : 0.0f;
        O = wmma4(ap, bv, O);
      }
    }
    // epilogue: heads.transpose(1,2,0,3) scatter -> Hm[b*G+g, h*8+d]
    if (n < 8) {
#pragma unroll
      for (int v = 0; v < 8; ++v) {
        int g = qbase + v + 8 * hi;
        Hm[(size_t)(b * kG + g) * kKD + h * kDH + n] = O[v] / lrow[v];
      }
    }
  }
}

// ---------------------------------------------------------------------------
// Kernel 3: out (16384x512) = Hm (16384x64) @ Wout^T (64x512) + bias
// ---------------------------------------------------------------------------
__global__ __launch_bounds__(256) void out_proj_kernel(
    const float* __restrict__ Hm, const float* __restrict__ Wo,
    const float* __restrict__ bias, float* __restrict__ out) {
  __shared__ __align__(16) float Hs[64 * S1];
  __shared__ __align__(16) float Wsr[128 * S1];
  const int t = threadIdx.x, lane = t & 31, w = t >> 5;
  const int n = lane & 15, hi = lane >> 4;
  const int row0 = blockIdx.x * 64;
  const int col0 = blockIdx.y * 128;
#pragma unroll
  for (int i = 0; i < 4; ++i) {
    int idx = t + i * 256;  // 0..1023
    int r = idx >> 4, c4 = (idx & 15) << 2;
    *(float4*)(&Hs[r * S1 + c4]) =
        *(const float4*)(Hm + (size_t)(row0 + r) * kKD + c4);
  }
#pragma unroll
  for (int i = 0; i < 8; ++i) {
    int idx = t + i * 256;  // 0..2047
    int r = idx >> 4, c4 = (idx & 15) << 2;
    *(float4*)(&Wsr[r * S1 + c4]) =
        *(const float4*)(Wo + (size_t)(col0 + r) * kKD + c4);
  }
  __syncthreads();
  const int rt = (w & 3) * 16;
  const int cb = (w >> 2) * 64;
  v8f acc[4];
#pragma unroll
  for (int i = 0; i < 4; ++i)
#pragma unroll
    for (int v = 0; v < 8; ++v) acc[i][v] = 0.0f;
  for (int ks = 0; ks < 16; ++ks) {
    const int kb = ks * 4 + 2 * hi;
    v2f a;
    a[0] = Hs[(rt + n) * S1 + kb + 0];
    a[1] = Hs[(rt + n) * S1 + kb + 1];
#pragma unroll
    for (int i = 0; i < 4; ++i) {
      v2f bf;
      bf[0] = Wsr[(cb + i * 16 + n) * S1 + kb + 0];
      bf[1] = Wsr[(cb + i * 16 + n) * S1 + kb + 1];
      acc[i] = wmma4(a, bf, acc[i]);
    }
  }
#pragma unroll
  for (int i = 0; i < 4; ++i) {
    const int cg = col0 + cb + i * 16 + n;
    const float bb = bias[cg];
#pragma unroll
    for (int v = 0; v < 8; ++v)
      out[(size_t)(row0 + rt + v + 8 * hi) * kE + cg] = acc[i][v] + bb;
  }
}

// ---------------------------------------------------------------------------
extern "C" void kernel_launch(void* const* d_in, const int* in_sizes, int n_in,
                              void* d_out, int out_size, void* d_ws,
                              size_t ws_size, hipStream_t stream) {
  const float* q = (const float*)d_in[0];
  const unsigned char* mask = (const unsigned char*)d_in[1];  // jnp bool_
  const float* Wq = (const float*)d_in[2];
  const float* Wk = (const float*)d_in[3];
  const float* Wv = (const float*)d_in[4];
  const float* Wo = (const float*)d_in[5];
  const float* Wb = (const float*)d_in[6];
  float* out = (float*)d_out;

  float* ws = (float*)d_ws;
  const size_t blk = (size_t)kM * kKD;  // 1,048,576 floats
  float* Qm = ws;
  float* Km = ws + blk;
  float* Vm = ws + 2 * blk;
  float* Hm = ws + 3 * blk;  // ~16.8 MB total workspace

  qkv_proj_kernel<<<dim3(128), dim3(256), 0, stream>>>(q, Wq, Wk, Wv, Qm, Km,
                                                       Vm);
  attn_kernel<<<dim3(128, 4), dim3(256), 0, stream>>>(Qm, Km, Vm, mask, Hm);
  out_proj_kernel<<<dim3(256, 4), dim3(256), 0, stream>>>(Hm, Wo, Wb, out);
}